// LinkPredictionDecoder_16037407883984
// MI455X (gfx1250) — compile-verified
//
#include <hip/hip_runtime.h>

// LinkPredictionDecoder on MI455X (gfx1250), fp32 WMMA path.
//
// z [50000,128] f32, edge_index [2,E] i64, W1 [256,128], b1 [128],
// W2 [128,64], b2 [64], W3 [64,1], b3 [1]  ->  scores [E] f32.
//
// Block = 128 threads (4 wave32), 32 edges:
//   phase 0: async gather (global_load_async_to_lds_b128) of z[src]||z[dst]
//   phase 1: GEMM1 [32,256]x[256,128] via v_wmma_f32_16x16x4_f32, bias+ReLU
//   phase 2: GEMM2 [32,128]x[128,64]  via WMMA, bias+ReLU
//   phase 3: GEMM3 [32,64]x[64,1] per-lane dot (tiny)
//
// W1/W2 are repacked once per launch into fragment-major order in d_ws so
// each k-step needs a single coalesced global_load_b64 per lane.

#define D_DIM   128
#define TWO_D   256
#define H1_DIM  128
#define H2_DIM  64

#define A_STRIDE  260   // 256 + 4 pad (stride%64 words = 4 -> conflict-free)
#define H_STRIDE  132   // 128 + 4 pad
#define H2_STRIDE 68    //  64 + 4 pad

#define W1_SLOTS (8 * 64 * 32)   // n-tiles * k-steps * lanes (v2f each)
#define W2_SLOTS (4 * 32 * 32)

typedef __attribute__((ext_vector_type(2))) float v2f;
typedef __attribute__((ext_vector_type(8))) float v8f;

static __device__ __forceinline__ v8f wmma_f32x4(v2f a, v2f b, v8f c) {
    // D = A(16x4 f32) * B(4x16 f32) + C(16x16 f32); full fp32 matrix path.
    return __builtin_amdgcn_wmma_f32_16x16x4_f32(
        /*neg_a=*/false, a, /*neg_b=*/false, b,
        /*c_mod=*/(short)0, c, /*reuse_a=*/false, /*reuse_b=*/false);
}

// Raw LDS offset of a generic pointer known to point into shared memory.
static __device__ __forceinline__ unsigned lds_addr(const void* p) {
    return (unsigned)(unsigned long long)
        (const __attribute__((address_space(3))) void*)p;
}

// ---- weight repack: fragment-major layout ---------------------------------
// W1 slot j = (n*64 + kstep)*32 + lane -> {W[kk][ncol], W[kk+1][ncol]},
// kk = kstep*4 + 2*(lane>>4), ncol = n*16 + (lane&15).  W2 likewise.
__global__ __launch_bounds__(256)
void pack_weights_kernel(const float* __restrict__ W1,
                         const float* __restrict__ W2,
                         float* __restrict__ ws)
{
    const int j = (int)blockIdx.x * 256 + (int)threadIdx.x;
    if (j < W1_SLOTS) {
        const int lane = j & 31;
        const int k0s  = (j >> 5) & 63;
        const int n    = j >> 11;
        const int kk   = k0s * 4 + ((lane >> 4) << 1);
        const int ncol = n * 16 + (lane & 15);
        ws[2 * j]     = W1[kk * H1_DIM + ncol];
        ws[2 * j + 1] = W1[(kk + 1) * H1_DIM + ncol];
    } else if (j < W1_SLOTS + W2_SLOTS) {
        const int i    = j - W1_SLOTS;
        const int lane = i & 31;
        const int k0s  = (i >> 5) & 31;
        const int n    = i >> 10;
        const int kk   = k0s * 4 + ((lane >> 4) << 1);
        const int ncol = n * 16 + (lane & 15);
        ws[2 * j]     = W2[kk * H2_DIM + ncol];
        ws[2 * j + 1] = W2[(kk + 1) * H2_DIM + ncol];
    }
}

template <bool PACKED>
__global__ __launch_bounds__(128)
void linkpred_decoder_kernel(const float* __restrict__ z,
                             const long long* __restrict__ eidx,   // [2,E]
                             const float* __restrict__ W1,
                             const float* __restrict__ b1,
                             const float* __restrict__ W2,
                             const float* __restrict__ b2,
                             const float* __restrict__ W3,
                             const float* __restrict__ b3,
                             const float* __restrict__ wp,         // packed
                             float* __restrict__ out,
                             int E)
{
    __shared__ float As[32 * A_STRIDE];   // edge features [32][256] (+pad)
    __shared__ float Hs[32 * H_STRIDE];   // h1 activations [32][128] (+pad)
    float* H2s = As;                      // h2 reuses As after GEMM1

    const int tid  = threadIdx.x;
    const int wave = tid >> 5;
    const int lane = tid & 31;
    const int half = lane >> 4;
    const int l16  = lane & 15;
    const int e0   = (int)blockIdx.x * 32;

    // warm the (L2-resident) weights into near caches
    if (PACKED) {
        for (int off = tid * 64; off < 2 * (W1_SLOTS + W2_SLOTS);
             off += 128 * 64)
            __builtin_prefetch(&wp[off], 0, 0);
    } else {
        for (int off = tid * 64; off < TWO_D * H1_DIM; off += 128 * 64)
            __builtin_prefetch(&W1[off], 0, 0);
        if (tid * 64 < H1_DIM * H2_DIM)
            __builtin_prefetch(&W2[tid * 64], 0, 0);
    }

    // ---- phase 0: async gather straight into LDS -------------------------
    // 32 edges x 2 sides x 32 lanes x 16B; per-lane b128 async copies.
    for (int idx = tid; idx < 32 * 64; idx += 128) {
        const int m    = idx >> 6;
        const int q    = idx & 63;
        const int side = q >> 5;
        const int c    = q & 31;               // float4 chunk in the row
        int e = e0 + m;
        if (e >= E) e = E - 1;                 // clamp (masked at store)
        const long long vi = eidx[(size_t)side * (size_t)E + (size_t)e];
        const float* gp = z + (size_t)vi * D_DIM + (size_t)(c * 4);
        const unsigned lo = lds_addr(&As[m * A_STRIDE + side * D_DIM + c * 4]);
        asm volatile("global_load_async_to_lds_b128 %0, %1, off"
                     :: "v"(lo), "v"(gp) : "memory");
    }
    asm volatile("s_wait_asynccnt 0x0" ::: "memory");
    __syncthreads();

    // ---- phase 1: GEMM1 + bias + ReLU ------------------------------------
    const v2f* Wp1 = (const v2f*)wp;
    const v2f* Wp2 = Wp1 + W1_SLOTS;
    for (int nn = 0; nn < 2; ++nn) {
        const int n    = wave * 2 + nn;
        const int ncol = n * 16 + l16;
        const float bias = b1[ncol];
        v8f acc0 = {bias, bias, bias, bias, bias, bias, bias, bias};
        v8f acc1 = acc0;

        for (int k0 = 0; k0 < TWO_D; k0 += 4) {
            v2f bf;
            if (PACKED) {
                bf = Wp1[(n * 64 + (k0 >> 2)) * 32 + lane];
            } else {
                const int kk = k0 + (half << 1);
                bf.x = W1[kk * H1_DIM + ncol];
                bf.y = W1[(kk + 1) * H1_DIM + ncol];
            }
            const int kk = k0 + (half << 1);
            v2f a0, a1;
            a0.x = As[l16 * A_STRIDE + kk];
            a0.y = As[l16 * A_STRIDE + kk + 1];
            a1.x = As[(l16 + 16) * A_STRIDE + kk];
            a1.y = As[(l16 + 16) * A_STRIDE + kk + 1];
            acc0 = wmma_f32x4(a0, bf, acc0);
            acc1 = wmma_f32x4(a1, bf, acc1);
        }
        #pragma unroll
        for (int r = 0; r < 8; ++r) {
            const int row = r + 8 * half;
            Hs[row * H_STRIDE + ncol]        = fmaxf(acc0[r], 0.0f);
            Hs[(row + 16) * H_STRIDE + ncol] = fmaxf(acc1[r], 0.0f);
        }
    }
    __syncthreads();

    // ---- phase 2: GEMM2 + bias + ReLU ------------------------------------
    {
        const int n    = wave;
        const int ncol = n * 16 + l16;
        const float bias = b2[ncol];
        v8f acc0 = {bias, bias, bias, bias, bias, bias, bias, bias};
        v8f acc1 = acc0;

        for (int k0 = 0; k0 < H1_DIM; k0 += 4) {
            v2f bf;
            if (PACKED) {
                bf = Wp2[(n * 32 + (k0 >> 2)) * 32 + lane];
            } else {
                const int kk = k0 + (half << 1);
                bf.x = W2[kk * H2_DIM + ncol];
                bf.y = W2[(kk + 1) * H2_DIM + ncol];
            }
            const int kk = k0 + (half << 1);
            v2f a0, a1;
            a0.x = Hs[l16 * H_STRIDE + kk];
            a0.y = Hs[l16 * H_STRIDE + kk + 1];
            a1.x = Hs[(l16 + 16) * H_STRIDE + kk];
            a1.y = Hs[(l16 + 16) * H_STRIDE + kk + 1];
            acc0 = wmma_f32x4(a0, bf, acc0);
            acc1 = wmma_f32x4(a1, bf, acc1);
        }
        #pragma unroll
        for (int r = 0; r < 8; ++r) {
            const int row = r + 8 * half;
            H2s[row * H2_STRIDE + ncol]        = fmaxf(acc0[r], 0.0f);
            H2s[(row + 16) * H2_STRIDE + ncol] = fmaxf(acc1[r], 0.0f);
        }
    }
    __syncthreads();

    // ---- phase 3: GEMM3 (64-dot) + store ---------------------------------
    if (tid < 32) {
        float sum = b3[0];
        #pragma unroll
        for (int k = 0; k < H2_DIM; ++k)
            sum += H2s[tid * H2_STRIDE + k] * W3[k];   // W3 scalar-broadcast
        const int e = e0 + tid;
        if (e < E) out[e] = sum;
    }
}

extern "C" void kernel_launch(void* const* d_in, const int* in_sizes, int n_in,
                              void* d_out, int out_size, void* d_ws, size_t ws_size,
                              hipStream_t stream) {
    const float*      z    = (const float*)d_in[0];
    const long long*  eidx = (const long long*)d_in[1];   // int64 [2,E]
    const float*      W1   = (const float*)d_in[2];
    const float*      b1   = (const float*)d_in[3];
    const float*      W2   = (const float*)d_in[4];
    const float*      b2   = (const float*)d_in[5];
    const float*      W3   = (const float*)d_in[6];
    const float*      b3   = (const float*)d_in[7];
    float*            out  = (float*)d_out;

    const int E = in_sizes[1] / 2;
    const int blocks = (E + 31) / 32;

    const size_t need = (size_t)(W1_SLOTS + W2_SLOTS) * 2 * sizeof(float);
    const bool packed = (d_ws != nullptr) && (ws_size >= need);

    if (packed) {
        const int pk = (W1_SLOTS + W2_SLOTS + 255) / 256;
        pack_weights_kernel<<<dim3(pk), dim3(256), 0, stream>>>(
            W1, W2, (float*)d_ws);
        linkpred_decoder_kernel<true><<<dim3(blocks), dim3(128), 0, stream>>>(
            z, eidx, W1, b1, W2, b2, W3, b3, (const float*)d_ws, out, E);
    } else {
        linkpred_decoder_kernel<false><<<dim3(blocks), dim3(128), 0, stream>>>(
            z, eidx, W1, b1, W2, b2, W3, b3, nullptr, out, E);
    }
}